// MultiresolutionHashEncoding_24936580120608
// MI455X (gfx1250) — compile-verified
//
#include <hip/hip_runtime.h>
#include <math.h>
#include <stdint.h>

// ---------------------------------------------------------------------------
// Multiresolution hash encoding (Instant-NGP style) for MI455X / gfx1250.
//   B=262144 points, D=3, L=16 levels, F=2 feats, T=2^19 table entries/level.
// Gather-bound workload: table (64MB) is L2-resident (192MB L2); WMMA does not
// apply (per-point 8x2 contraction, no shared operand). CDNA5 features used:
//   - TDM tensor_load_to_lds   : async DMA of the x tile into LDS
//   - TDM tensor_store_from_lds: coalesced 32KB output tile store per block
//   - s_wait_tensorcnt for TDM completion
// Toolchain: amdgpu-toolchain (clang-23) -> 6-arg TDM builtins.
// ---------------------------------------------------------------------------

#define T_SIZE   524288u        // 2^19 hash-table entries per level
#define NLEV     16
#define NFEAT    2
#define PTSB     256            // points per block
#define NPTS     262144         // B
#define PRIME1   2654435761u
#define PRIME2   805459861u

typedef unsigned int u32x4 __attribute__((ext_vector_type(4)));
typedef int          i32x4 __attribute__((ext_vector_type(4)));
typedef int          i32x8 __attribute__((ext_vector_type(8)));

struct ResArrF { float r[NLEV]; };   // resolutions pre-converted to f32 (all <= 512, exact)

// ---- Tensor DMA descriptor builders (CDNA5 ISA ch.8, D# groups 0/1) -------
// Group 0 (128b): [1:0]=count=1, [63:32]=lds_addr, [120:64]=global_addr,
//                 [127:126]=type=2 ("image").
static __device__ __forceinline__ u32x4 tdm_g0(uint32_t lds_addr, uint64_t gaddr) {
  u32x4 g;
  g[0] = 1u;                                                  // count=1, user mode
  g[1] = lds_addr;                                            // LDS byte address
  g[2] = (uint32_t)gaddr;                                     // global_addr[31:0]
  g[3] = (uint32_t)((gaddr >> 32) & 0x1FFFFFFu) | (2u << 30); // [56:32] | type=2
  return g;
}

// Group 1 (256b) describing a simple 1-D contiguous copy of `tile_n` dwords
// out of a tensor of `tensor_n` dwords. data_size=2 => 4-byte elements.
static __device__ __forceinline__ i32x8 tdm_g1_1d(uint32_t tile_n, uint32_t tensor_n) {
  i32x8 g;
  g[0] = (int)(2u << 16);                            // wg_mask=0, data_size=4B, no flags
  g[1] = (int)((tensor_n & 0xFFFFu) << 16);          // atomic_bar_addr=0 | tensor_dim0 lo16
  g[2] = (int)(((tensor_n >> 16) & 0xFFFFu)          // tensor_dim0 hi16
               | (1u << 16));                        // tensor_dim1 lo16 = 1
  g[3] = (int)(tile_n << 16);                        // tensor_dim1 hi16=0 | tile_dim0
  g[4] = 0;                                          // tile_dim1=0 (unused), tile_dim2=0
  g[5] = (int)tensor_n;                              // tensor_dim0_stride lo32
  g[6] = 0;                                          // stride hi16 | dim1_stride lo16
  g[7] = 0;                                          // dim1_stride hi32
  return g;
}

__global__ __launch_bounds__(PTSB)
void MultiresolutionHashEncoding_24936580120608_kernel(
    const float* __restrict__ x,       // (B, 3)
    const float* __restrict__ tables,  // (L, T, 2)
    float* __restrict__ out,           // (B, L*2)
    ResArrF res) {
  __shared__ float xs[PTSB * 3];              // 3 KB input tile
  __shared__ float outs[PTSB * NLEV * NFEAT]; // 32 KB output tile

  const int tid = threadIdx.x;
  const int b0  = blockIdx.x * PTSB;

  // ---- Stage x tile into LDS via Tensor Data Mover ------------------------
#if __has_builtin(__builtin_amdgcn_tensor_load_to_lds)
  if (tid == 0) {
    u32x4 g0 = tdm_g0((uint32_t)(uintptr_t)&xs[0],
                      (uint64_t)(uintptr_t)(x + (size_t)b0 * 3));
    i32x8 g1 = tdm_g1_1d(PTSB * 3, NPTS * 3u);
    i32x4 z4 = {0, 0, 0, 0};
    i32x8 z8 = {0, 0, 0, 0, 0, 0, 0, 0};
    __builtin_amdgcn_tensor_load_to_lds(g0, g1, z4, z4, z8, 0);
    __builtin_amdgcn_s_wait_tensorcnt(0);   // data resident before barrier release
  }
#else
  for (int i = tid; i < PTSB * 3; i += PTSB) xs[i] = x[(size_t)b0 * 3 + i];
#endif
  __syncthreads();

  const float px = xs[tid * 3 + 0];
  const float py = xs[tid * 3 + 1];
  const float pz = xs[tid * 3 + 2];

  // ---- Per-level hash + gather + trilinear blend --------------------------
  // T = 2^19 and all grid coords are >= 0, so the reference's int64
  // (xor of prime products) mod T reduces exactly to u32 arithmetic.
  // unroll 4 => 32 global_load_b64 gathers in flight per thread for L2
  // latency hiding (the kernel's real bottleneck).
#pragma unroll 4
  for (int l = 0; l < NLEV; ++l) {
    const float rf = res.r[l];
    const float sx = px * rf, sy = py * rf, sz = pz * rf;
    const float fx = floorf(sx), fy = floorf(sy), fz = floorf(sz);
    const float tx = sx - fx, ty = sy - fy, tz = sz - fz;

    const uint32_t h0 = (uint32_t)(int)fx;
    const uint32_t h1 = (uint32_t)(int)fy * PRIME1;
    const uint32_t h2 = (uint32_t)(int)fz * PRIME2;
    const uint32_t e0[2] = {h0, h0 + 1u};
    const uint32_t e1[2] = {h1, h1 + PRIME1};
    const uint32_t e2[2] = {h2, h2 + PRIME2};
    const float    wx[2] = {1.0f - tx, tx};
    const float    wy[2] = {1.0f - ty, ty};
    const float    wz[2] = {1.0f - tz, tz};

    const float2* tab = (const float2*)tables + (size_t)l * T_SIZE;

    float a0 = 0.0f, a1 = 0.0f;
#pragma unroll
    for (int c = 0; c < 8; ++c) {           // 8 corners, all loads issued together
      const int c0 = c & 1, c1 = (c >> 1) & 1, c2 = c >> 2;
      const uint32_t idx = (e0[c0] ^ e1[c1] ^ e2[c2]) & (T_SIZE - 1u);
      const float2 f = tab[idx];
      const float  w = wx[c0] * wy[c1] * wz[c2];
      a0 = fmaf(w, f.x, a0);
      a1 = fmaf(w, f.y, a1);
    }
    outs[tid * (NLEV * NFEAT) + l * NFEAT + 0] = a0;
    outs[tid * (NLEV * NFEAT) + l * NFEAT + 1] = a1;
  }
  __syncthreads();   // all DS writes visible before TDM reads LDS

  // ---- Store the contiguous 32KB output tile via TDM ----------------------
#if __has_builtin(__builtin_amdgcn_tensor_store_from_lds)
  if (tid == 0) {
    u32x4 g0 = tdm_g0((uint32_t)(uintptr_t)&outs[0],
                      (uint64_t)(uintptr_t)(out + (size_t)b0 * (NLEV * NFEAT)));
    i32x8 g1 = tdm_g1_1d(PTSB * NLEV * NFEAT, NPTS * (NLEV * NFEAT));
    i32x4 z4 = {0, 0, 0, 0};
    i32x8 z8 = {0, 0, 0, 0, 0, 0, 0, 0};
    __builtin_amdgcn_tensor_store_from_lds(g0, g1, z4, z4, z8, 0);
    __builtin_amdgcn_s_wait_tensorcnt(0);   // keep LDS alive until DMA completes
  }
#else
  for (int i = tid; i < PTSB * NLEV * NFEAT; i += PTSB)
    out[(size_t)b0 * (NLEV * NFEAT) + i] = outs[i];
#endif
}

extern "C" void kernel_launch(void* const* d_in, const int* in_sizes, int n_in,
                              void* d_out, int out_size, void* d_ws, size_t ws_size,
                              hipStream_t stream) {
  const float* x      = (const float*)d_in[0];
  const float* tables = (const float*)d_in[1];
  float*       out    = (float*)d_out;

  // Resolutions: floor(16 * b^l), b = exp((ln 512 - ln 16)/15)  (host double,
  // matching the numpy computation path), then exact f32 conversion.
  ResArrF res;
  const double b = exp((log(512.0) - log(16.0)) / 15.0);
  for (int i = 0; i < NLEV; ++i)
    res.r[i] = (float)(int)floor(16.0 * pow(b, (double)i));

  MultiresolutionHashEncoding_24936580120608_kernel<<<NPTS / PTSB, PTSB, 0, stream>>>(
      x, tables, out, res);
}